// GraphSAGE_82592221102739
// MI455X (gfx1250) — compile-verified
//
#include <hip/hip_runtime.h>
#include <hip/hip_bf16.h>
#include <stdint.h>

// ---------------------------------------------------------------------------
// GraphSAGE (2-layer, mean aggregation) for MI455X / gfx1250.
//   N=50000 nodes, E=800000 edges, dims 128->128->128, fp32 throughout.
// Memory-bound on edge gather/scatter (~1.6 GB irregular traffic); GEMMs are
// tiny (6.6 GFLOP) so we keep full fp32 fidelity via V_WMMA_F32_16X16X4_F32.
// ---------------------------------------------------------------------------

typedef __attribute__((ext_vector_type(2))) float v2f;
typedef __attribute__((ext_vector_type(8))) float v8f;

#define DIM 128
#define LDS_STRIDE 132   // 128 + 4 pad floats: rows land on distinct bank sets

// ---------------------------------------------------------------------------
// Degree accumulation: one thread per edge, atomic +1.0f at dst.
// ---------------------------------------------------------------------------
__global__ void deg_kernel(const long long* __restrict__ dst,
                           float* __restrict__ deg, int E) {
  int e = blockIdx.x * blockDim.x + threadIdx.x;
  if (e < E) {
    atomicAdd(&deg[dst[e]], 1.0f);
  }
}

// deg -> 1/max(deg,1), in place.
__global__ void deg_inv_kernel(float* __restrict__ deg, int N) {
  int i = blockIdx.x * blockDim.x + threadIdx.x;
  if (i < N) {
    float d = deg[i];
    deg[i] = 1.0f / (d > 1.0f ? d : 1.0f);
  }
}

// ---------------------------------------------------------------------------
// Edge scatter: wave-per-edge. 32 lanes x float4 = full 128-float feature row.
// Gather from feat[src], scatter-add into msg[dst] (global_atomic_add_f32,
// non-returning -> STOREcnt only).
// ---------------------------------------------------------------------------
__global__ void scatter_kernel(const float* __restrict__ feat,
                               const long long* __restrict__ src,
                               const long long* __restrict__ dst,
                               float* __restrict__ msg, int E) {
  long long tid = (long long)blockIdx.x * blockDim.x + threadIdx.x;
  long long e = tid >> 5;          // edge id (one wave32 per edge)
  int c = (int)(tid & 31);         // float4 chunk within the 128-float row
  if (e >= E) return;
  long long s = src[e];
  long long d = dst[e];
  float4 v = ((const float4*)(feat + s * DIM))[c];
  float* p = msg + d * DIM + (long long)c * 4;
  atomicAdd(p + 0, v.x);
  atomicAdd(p + 1, v.y);
  atomicAdd(p + 2, v.z);
  atomicAdd(p + 3, v.w);
}

// ---------------------------------------------------------------------------
// Fused SAGE layer GEMM:
//   out = relu( (msg * deg_inv) @ Wl^T + bl + h @ Wr^T )
// Block = 256 threads (8 waves). Block owns a 16-row node tile; each wave owns
// one 16-wide output column tile. Activations staged in LDS (padded stride),
// weights stream from global (64 KB each -> L2 resident). 64 unrolled
// v_wmma_f32_16x16x4_f32 per wave.
//
// Fragment layouts per CDNA5 ISA 7.12.2:
//   A 16x4 f32 : lane m = M (m = lane&15), kk = 2*(lane>>4); v0=K(kk), v1=K(kk+1)
//   B 4x16 f32 : lane n = N (n = lane&15), kk = 2*(lane>>4); v0=K(kk), v1=K(kk+1)
//   C/D 16x16  : VGPR v, lanes 0-15 -> M=v, lanes 16-31 -> M=v+8; N = lane&15
// ---------------------------------------------------------------------------
__global__ __launch_bounds__(256) void sage_gemm_kernel(
    const float* __restrict__ msg, const float* __restrict__ deg_inv,
    const float* __restrict__ h,   const float* __restrict__ Wl,
    const float* __restrict__ bl,  const float* __restrict__ Wr,
    float* __restrict__ out, int N) {
  __shared__ float Atile[16 * LDS_STRIDE];  // agg = msg * deg_inv
  __shared__ float Htile[16 * LDS_STRIDE];  // h (self features)

  const int node0 = blockIdx.x * 16;
  const int tid = threadIdx.x;

  // ---- Stage 16x128 agg tile and 16x128 h tile into LDS (float4 traffic).
  for (int i = tid; i < 16 * (DIM / 4); i += 256) {
    int r = i >> 5;          // row in tile
    int c4 = i & 31;         // float4 chunk
    int node = node0 + r;
    int nclamp = node < N ? node : (N - 1);   // keep EXEC full, clamp loads
    float4 m4 = ((const float4*)(msg + (size_t)nclamp * DIM))[c4];
    float di = deg_inv[nclamp];
    m4.x *= di; m4.y *= di; m4.z *= di; m4.w *= di;
    *(float4*)(&Atile[r * LDS_STRIDE + c4 * 4]) = m4;
    float4 h4 = ((const float4*)(h + (size_t)nclamp * DIM))[c4];
    *(float4*)(&Htile[r * LDS_STRIDE + c4 * 4]) = h4;
  }
  __syncthreads();

  const int wave = tid >> 5;        // 0..7 -> output column tile
  const int lane = tid & 31;
  const int n0 = wave * 16;
  const int mn = lane & 15;         // doubles as M (for A) and N (for B/C/D)
  const int kk = (lane >> 4) << 1;  // 0 or 2: K sub-offset per lane half

  v8f acc = {0.f, 0.f, 0.f, 0.f, 0.f, 0.f, 0.f, 0.f};

  const float* wlrow = Wl + (size_t)(n0 + mn) * DIM;  // B[k][n] = Wl[n][k]
  const float* wrrow = Wr + (size_t)(n0 + mn) * DIM;

  // ---- agg @ Wl^T : 32 x K=4 steps
#pragma unroll
  for (int kb = 0; kb < DIM; kb += 4) {
    v2f a = *(const v2f*)(&Atile[mn * LDS_STRIDE + kb + kk]);  // ds_load_b64
    v2f b = *(const v2f*)(&wlrow[kb + kk]);                    // global_load_b64
    acc = __builtin_amdgcn_wmma_f32_16x16x4_f32(
        false, a, false, b, (short)0, acc, false, false);
  }
  // ---- h @ Wr^T : 32 x K=4 steps
#pragma unroll
  for (int kb = 0; kb < DIM; kb += 4) {
    v2f a = *(const v2f*)(&Htile[mn * LDS_STRIDE + kb + kk]);
    v2f b = *(const v2f*)(&wrrow[kb + kk]);
    acc = __builtin_amdgcn_wmma_f32_16x16x4_f32(
        false, a, false, b, (short)0, acc, false, false);
  }

  // ---- Epilogue: bias + ReLU, scatter C fragment to out.
  const float bias = bl[n0 + mn];
  const int rbase = (lane >> 4) * 8;   // lanes 16-31 hold M = v + 8
#pragma unroll
  for (int v = 0; v < 8; ++v) {
    int row = node0 + rbase + v;
    if (row < N) {
      float val = acc[v] + bias;
      out[(size_t)row * DIM + n0 + mn] = val > 0.f ? val : 0.f;
    }
  }
}

// ---------------------------------------------------------------------------
// Launch: deg -> deg_inv -> scatter -> GEMM (L1) -> scatter -> GEMM (L2)
// Workspace: [deg/deg_inv: N][msg: N*128][h1: N*128]  ~51.4 MB fp32
// ---------------------------------------------------------------------------
extern "C" void kernel_launch(void* const* d_in, const int* in_sizes, int n_in,
                              void* d_out, int out_size, void* d_ws, size_t ws_size,
                              hipStream_t stream) {
  const float* x   = (const float*)d_in[0];
  const float* Wl1 = (const float*)d_in[1];
  const float* bl1 = (const float*)d_in[2];
  const float* Wr1 = (const float*)d_in[3];
  const float* Wl2 = (const float*)d_in[4];
  const float* bl2 = (const float*)d_in[5];
  const float* Wr2 = (const float*)d_in[6];
  const long long* edge_index = (const long long*)d_in[7];

  const int N = in_sizes[0] / DIM;
  const int E = in_sizes[7] / 2;
  const long long* src = edge_index;
  const long long* dst = edge_index + E;

  float* deg = (float*)d_ws;                         // N floats (becomes deg_inv)
  size_t degpad = ((size_t)N + 255) & ~(size_t)255;
  float* msg = deg + degpad;                         // N*128 floats
  float* h1  = msg + (size_t)N * DIM;                // N*128 floats
  float* out = (float*)d_out;

  const size_t msg_bytes = (size_t)N * DIM * sizeof(float);

  // degree + inverse
  hipMemsetAsync(deg, 0, (size_t)N * sizeof(float), stream);
  hipMemsetAsync(msg, 0, msg_bytes, stream);
  deg_kernel<<<(E + 255) / 256, 256, 0, stream>>>(dst, deg, E);
  deg_inv_kernel<<<(N + 255) / 256, 256, 0, stream>>>(deg, N);

  const long long scatter_threads = (long long)E * 32;
  const int scatter_blocks = (int)((scatter_threads + 255) / 256);
  const int gemm_blocks = (N + 15) / 16;

  // Layer 1
  scatter_kernel<<<scatter_blocks, 256, 0, stream>>>(x, src, dst, msg, E);
  sage_gemm_kernel<<<gemm_blocks, 256, 0, stream>>>(msg, deg, x, Wl1, bl1, Wr1,
                                                    h1, N);
  // Layer 2
  hipMemsetAsync(msg, 0, msg_bytes, stream);
  scatter_kernel<<<scatter_blocks, 256, 0, stream>>>(h1, src, dst, msg, E);
  sage_gemm_kernel<<<gemm_blocks, 256, 0, stream>>>(msg, deg, h1, Wl2, bl2, Wr2,
                                                    out, N);
}